// DenseRelationDistill_35459249996641
// MI455X (gfx1250) — compile-verified
//
#include <hip/hip_runtime.h>

typedef float v2f __attribute__((ext_vector_type(2)));
typedef float v8f __attribute__((ext_vector_type(8)));

__device__ __forceinline__ v8f wmma_f32(v2f a, v2f b, v8f c) {
  // V_WMMA_F32_16X16X4_F32 : D(16x16 f32) = A(16x4) * B(4x16) + C
  return __builtin_amdgcn_wmma_f32_16x16x4_f32(false, a, false, b, (short)0, c, false, false);
}

// ---------------------------------------------------------------------------
// 1. Bilinear downsample f[4,256,H,W] -> fs[4,256,16,16] (align_corners=True)
// ---------------------------------------------------------------------------
__global__ __launch_bounds__(256) void k_downsample(const float* __restrict__ f,
                                                    float* __restrict__ fs,
                                                    int H, int W) {
  int idx = blockIdx.x * 256 + threadIdx.x;        // exactly 4*256*256 threads
  int x = idx & 15, y = (idx >> 4) & 15;
  int c = (idx >> 8) & 255;
  int b = idx >> 16;
  float sy = (float)y * (float)(H - 1) * (1.0f / 15.0f);
  float sx = (float)x * (float)(W - 1) * (1.0f / 15.0f);
  int y0 = (int)floorf(sy), x0 = (int)floorf(sx);
  int y1 = min(y0 + 1, H - 1), x1 = min(x0 + 1, W - 1);
  float wy = sy - (float)y0, wx = sx - (float)x0;
  const float* fc = f + (size_t)(b * 256 + c) * H * W;
  float v00 = fc[y0 * W + x0], v01 = fc[y0 * W + x1];
  float v10 = fc[y1 * W + x0], v11 = fc[y1 * W + x1];
  fs[idx] = (v00 * (1.f - wx) + v01 * wx) * (1.f - wy) +
            (v10 * (1.f - wx) + v11 * wx) * wy;
}

// ---------------------------------------------------------------------------
// 2a. Conv weight transpose: w[OC,256,3,3] -> wt4[(r*256+c)*4 + s][OC], s==3 -> 0
// ---------------------------------------------------------------------------
__global__ __launch_bounds__(256) void k_tr_convw(const float* __restrict__ w,
                                                  float* __restrict__ wt4, int OC) {
  int idx = blockIdx.x * 256 + threadIdx.x;
  if (idx >= OC * 768) return;
  int r  = idx % 3;
  int c  = (idx / 3) % 256;
  int oc = idx / 768;
  int g  = r * 256 + c;
  const float* src = w + (((size_t)oc * 256 + c) * 3 + r) * 3;
  float* dst = wt4 + (size_t)(g * 4) * OC + oc;
  dst[0]              = src[0];
  dst[(size_t)OC]     = src[1];
  dst[(size_t)2 * OC] = src[2];
  dst[(size_t)3 * OC] = 0.f;
}

// 2b. Combine weight transpose: w[256,512] -> wt[512][256]
__global__ __launch_bounds__(256) void k_tr_combw(const float* __restrict__ w,
                                                  float* __restrict__ wt) {
  int idx = blockIdx.x * 256 + threadIdx.x;        // 131072 threads
  int k  = idx & 511;
  int oc = idx >> 9;
  wt[(size_t)k * 256 + oc] = w[idx];
}

// 2c. val_t transpose: v[20,128,256] -> vT[20,256,128]
__global__ __launch_bounds__(256) void k_tr_valt(const float* __restrict__ v,
                                                 float* __restrict__ vT) {
  int idx = blockIdx.x * 256 + threadIdx.x;        // 655360 threads
  int q = idx & 255;
  int c = (idx >> 8) & 127;
  int n = idx >> 15;
  vT[((size_t)n * 256 + q) * 128 + c] = v[idx];
}

// ---------------------------------------------------------------------------
// 3. Conv3x3 (pad=1) on 16x16 images as WMMA GEMM.
//    One wave per (batch, row-pair y0/y0+1, 16-out-channel tile); the weight
//    A-fragment is shared by both rows (2 wmma per weight fetch).
//    Halo handling: clamped unconditional loads + v_cndmask (no EXEC branches).
// ---------------------------------------------------------------------------
template <int OC>
__global__ __launch_bounds__(32) void k_conv3x3_wmma(const float* __restrict__ in,
    const float* __restrict__ wt4, const float* __restrict__ bias,
    float* __restrict__ out) {
  const int lane = threadIdx.x;
  const int hf = lane >> 4, l = lane & 15;
  const int y0 = (blockIdx.x & 7) << 1, b = blockIdx.x >> 3;
  const int oc0 = blockIdx.y << 4;

  v8f acc0, acc1;
#pragma unroll
  for (int j = 0; j < 8; ++j) {
    float bv = bias[oc0 + j + hf * 8];
    acc0[j] = bv; acc1[j] = bv;
  }

  const float* inb = in + (size_t)b * 256 * 256;
  const int  xi  = l + hf * 2 - 1;              // s=0 -> x-1 ; s=2 -> x+1
  const int  xc  = min(max(xi, 0), 15);         // clamped (always-valid) column
  const bool xok = (xi >= 0) && (xi <= 15);

  for (int r = 0; r < 3; ++r) {
    int  yyA = y0 + r - 1;                      // input row for out-row y0
    int  yyB = yyA + 1;                         // input row for out-row y0+1
    bool okA = (yyA >= 0);                      // yyA <= 15 always (y0<=14, r<=2 -> yyA<=15)
    bool okB = (yyB <= 15);                     // yyB >= 0 always
    const float* rowA = inb + (size_t)min(max(yyA, 0), 15) * 16;
    const float* rowB = inb + (size_t)min(yyB, 15) * 16;
    const float* wr = wt4 + (size_t)(r * 1024 + hf * 2) * OC + oc0 + l;
#pragma unroll 2
    for (int c = 0; c < 256; ++c) {
      v2f a;
      a.x = wr[0];                              // A[m=oc][k]   (coalesced)
      a.y = wr[OC];
      wr += 4 * OC;
      const float* pA = rowA + c * 256;
      const float* pB = rowB + c * 256;
      float vA0 = pA[xc], vA1 = pA[l];          // unconditional clamped loads
      float vB0 = pB[xc], vB1 = pB[l];
      v2f bfA, bfB;
      bfA.x = (okA && xok) ? vA0 : 0.f;         // cndmask, no exec branching
      bfA.y = (okA && !hf) ? vA1 : 0.f;         // s=1 ; s=3 is the zero pad
      bfB.x = (okB && xok) ? vB0 : 0.f;
      bfB.y = (okB && !hf) ? vB1 : 0.f;
      acc0 = wmma_f32(a, bfA, acc0);
      acc1 = wmma_f32(a, bfB, acc1);
    }
  }
  float* ob = out + (size_t)b * OC * 256 + y0 * 16;
#pragma unroll
  for (int j = 0; j < 8; ++j) {
    float* o = ob + (size_t)(oc0 + j + hf * 8) * 256 + l;
    o[0]  = acc0[j];                            // row y0   (coalesced over x)
    o[16] = acc1[j];                            // row y0+1
  }
}

// ---------------------------------------------------------------------------
// 4. Attention logits: P[b,n,q,t] = 0.5 * sum_k kq[b,k,q] * key_t[n,k,t]
//    One wave per 16q x 16t tile, K=32.
// ---------------------------------------------------------------------------
__global__ __launch_bounds__(32) void k_attn_logits(const float* __restrict__ kq,
    const float* __restrict__ keyt, float* __restrict__ P) {
  const int lane = threadIdx.x, hf = lane >> 4, l = lane & 15;
  const int b = blockIdx.x, n = blockIdx.y;
  const int q0 = (blockIdx.z >> 4) << 4, t0 = (blockIdx.z & 15) << 4;
  const float* ap = kq + (size_t)b * 32 * 256 + (hf * 2) * 256 + q0 + l;
  const float* bp = keyt + (size_t)n * 32 * 256 + (hf * 2) * 256 + t0 + l;
  v8f acc = {};
#pragma unroll
  for (int k0 = 0; k0 < 32; k0 += 4) {
    v2f a, bf;
    a.x = ap[k0 * 256];
    a.y = ap[(k0 + 1) * 256];
    bf.x = bp[k0 * 256];
    bf.y = bp[(k0 + 1) * 256];
    acc = wmma_f32(a, bf, acc);
  }
  float* Pb = P + (size_t)(b * 20 + n) * 65536;
#pragma unroll
  for (int j = 0; j < 8; ++j)
    Pb[(size_t)(q0 + j + hf * 8) * 256 + t0 + l] = 0.5f * acc[j];
}

// ---------------------------------------------------------------------------
// 5. Softmax over q (axis 2 of [b,n,q,t]). Block = 256 threads = 16t x 16q-groups,
//    16 q-values per thread in registers; LDS tree for max & sum.
// ---------------------------------------------------------------------------
__global__ __launch_bounds__(256) void k_softmax_q(float* __restrict__ P) {
  const int bn = blockIdx.x;
  const int t0 = blockIdx.y << 4;
  const int tt = threadIdx.x & 15, tq = threadIdx.x >> 4;
  float* base = P + (size_t)bn * 65536 + t0 + tt;
  float vals[16];
  float m = -3.4e38f;
#pragma unroll
  for (int i = 0; i < 16; ++i) {
    vals[i] = base[(size_t)(tq * 16 + i) * 256];
    m = fmaxf(m, vals[i]);
  }
  __shared__ float red[16][17];
  red[tq][tt] = m;
  __syncthreads();
  float gm = -3.4e38f;
#pragma unroll
  for (int j = 0; j < 16; ++j) gm = fmaxf(gm, red[j][tt]);
  float s = 0.f;
#pragma unroll
  for (int i = 0; i < 16; ++i) { vals[i] = __expf(vals[i] - gm); s += vals[i]; }
  __syncthreads();
  red[tq][tt] = s;
  __syncthreads();
  float gs = 0.f;
#pragma unroll
  for (int j = 0; j < 16; ++j) gs += red[j][tt];
  float inv = 1.f / gs;
#pragma unroll
  for (int i = 0; i < 16; ++i)
    base[(size_t)(tq * 16 + i) * 256] = vals[i] * inv;
}

// ---------------------------------------------------------------------------
// 6. vt[b,c,t] = sum_n sum_q val_t[n,c,q] * P[b,n,q,t]
//    One wave per 16c x 16t tile; K=256 per class, 20 classes -> 1280 wmma.
// ---------------------------------------------------------------------------
__global__ __launch_bounds__(32) void k_attn_apply(const float* __restrict__ valtT,
    const float* __restrict__ P, float* __restrict__ vt) {
  const int lane = threadIdx.x, hf = lane >> 4, l = lane & 15;
  const int b = blockIdx.x;
  const int c0 = blockIdx.y << 4, t0 = blockIdx.z << 4;
  v8f acc = {};
  for (int n = 0; n < 20; ++n) {
    const float* ap = valtT + (size_t)n * 256 * 128 + (hf * 2) * 128 + c0 + l;
    const float* bp = P + (size_t)(b * 20 + n) * 65536 + (hf * 2) * 256 + t0 + l;
    if (n + 1 < 20)
      __builtin_prefetch(P + (size_t)(b * 20 + n + 1) * 65536 + t0, 0, 0);
#pragma unroll 4
    for (int k0 = 0; k0 < 256; k0 += 4) {
      v2f a, bf;
      a.x = ap[(size_t)k0 * 128];
      a.y = ap[(size_t)(k0 + 1) * 128];
      bf.x = bp[(size_t)k0 * 256];
      bf.y = bp[(size_t)(k0 + 1) * 256];
      acc = wmma_f32(a, bf, acc);
    }
  }
  float* ob = vt + (size_t)b * 128 * 256;
#pragma unroll
  for (int j = 0; j < 8; ++j)
    ob[(size_t)(c0 + j + hf * 8) * 256 + t0 + l] = acc[j];
}

// ---------------------------------------------------------------------------
// 7. Fused combine: out[b,oc,p] = cb[oc] + sum_{k<512} W[oc,k] * X[k,p]
//    X[k<256]  = f channel k at pixel p
//    X[k>=256] = BN(bilinear-upsample(cat(20*vq, vt)))[k-256] at pixel p
//    Block = 256 threads, 32-pixel tile: two 512x16 LDS activation arrays
//    (stride-16 rows -> conflict-free B reads), dual accumulators so each
//    weight A-fragment feeds two wmma.
// ---------------------------------------------------------------------------
__global__ __launch_bounds__(256) void k_combine(const float* __restrict__ f,
    const float* __restrict__ wTc, const float* __restrict__ cb,
    const float* __restrict__ vq, const float* __restrict__ vt,
    const float* __restrict__ bng, const float* __restrict__ bnb,
    const float* __restrict__ bnm, const float* __restrict__ bnv,
    float* __restrict__ out, int H, int W) {
  __shared__ float xts[2 * 512 * 16];                 // 64 KB: xt0 | xt1
  const int HW = H * W;
  const int ntile = (HW + 31) >> 5;                   // 32-pixel tiles (ceil)
  const int tile = blockIdx.x % ntile;
  const int b = blockIdx.x / ntile;
  const int tid = threadIdx.x;
  const int xl = tid & 31, c0 = tid >> 5;             // 32 pixels x 8 c-groups
  const int p = min(tile * 32 + xl, HW - 1);          // clamp ragged tail
  const int y = p / W, x = p % W;
  const int sbase = (xl >= 16 ? 512 * 16 : 0) + (xl & 15);

  // stage f channels (coalesced 128B per wave)
  const float* fb = f + (size_t)b * 256 * HW + p;
#pragma unroll
  for (int i = 0; i < 32; ++i) {
    int c = i * 8 + c0;
    xts[sbase + c * 16] = fb[(size_t)c * HW];
  }
  // stage "final" channels: bilinear up from 16x16 + batchnorm
  float sy = (float)y * (15.0f / (float)(H - 1));
  float sx = (float)x * (15.0f / (float)(W - 1));
  int y0 = (int)floorf(sy), x0 = (int)floorf(sx);
  int y1 = min(y0 + 1, 15), x1 = min(x0 + 1, 15);
  float wy = sy - (float)y0, wx = sx - (float)x0;
  int p00 = y0 * 16 + x0, p01 = y0 * 16 + x1;
  int p10 = y1 * 16 + x0, p11 = y1 * 16 + x1;
  const float* vqb = vq + (size_t)b * 128 * 256;
  const float* vtb = vt + (size_t)b * 128 * 256;
#pragma unroll
  for (int i = 0; i < 32; ++i) {
    int c = i * 8 + c0;
    const float* src = (c < 128) ? (vqb + (size_t)c * 256)
                                 : (vtb + (size_t)(c - 128) * 256);
    float scl = (c < 128) ? 20.0f : 1.0f;             // ncls * vq
    float v00 = src[p00], v01 = src[p01], v10 = src[p10], v11 = src[p11];
    float v = ((v00 * (1.f - wx) + v01 * wx) * (1.f - wy) +
               (v10 * (1.f - wx) + v11 * wx) * wy) * scl;
    float bscale = bng[c] * rsqrtf(bnv[c] + 1e-5f);
    xts[sbase + (256 + c) * 16] = (v - bnm[c]) * bscale + bnb[c];
  }
  __syncthreads();

  const int wave = tid >> 5, lane = tid & 31, hf = lane >> 4, l = lane & 15;
  const int pbase = tile * 32;
  float* ob = out + (size_t)b * 256 * HW + pbase;
  const bool ok0 = (pbase + l) < HW, ok1 = (pbase + 16 + l) < HW;
#pragma unroll 1
  for (int rep = 0; rep < 2; ++rep) {
    int oc0 = (wave + rep * 8) << 4;
    v8f acc0, acc1;
#pragma unroll
    for (int j = 0; j < 8; ++j) {
      float bv = cb[oc0 + j + hf * 8];
      acc0[j] = bv; acc1[j] = bv;
    }
    const float* wp = wTc + (size_t)(hf * 2) * 256 + oc0 + l;
    const float* s0 = xts + hf * 32 + l;              // rows kb = k0 + 2*hf
    const float* s1 = s0 + 512 * 16;
#pragma unroll 2
    for (int k0 = 0; k0 < 512; k0 += 4) {
      v2f a, b0, b1;
      a.x = wp[(size_t)k0 * 256];                     // coalesced weight loads
      a.y = wp[(size_t)(k0 + 1) * 256];
      b0.x = s0[k0 * 16];  b0.y = s0[k0 * 16 + 16];
      b1.x = s1[k0 * 16];  b1.y = s1[k0 * 16 + 16];
      acc0 = wmma_f32(a, b0, acc0);                   // pixels 0..15
      acc1 = wmma_f32(a, b1, acc1);                   // pixels 16..31
    }
#pragma unroll
    for (int j = 0; j < 8; ++j) {
      int oc = oc0 + j + hf * 8;
      float* o = ob + (size_t)oc * HW;
      if (ok0) o[l] = acc0[j];                        // coalesced 64B stores
      if (ok1) o[16 + l] = acc1[j];
    }
  }
}

// ---------------------------------------------------------------------------
extern "C" void kernel_launch(void* const* d_in, const int* in_sizes, int n_in,
                              void* d_out, int out_size, void* d_ws, size_t ws_size,
                              hipStream_t stream) {
  (void)in_sizes; (void)n_in; (void)out_size; (void)ws_size;
  const float* f[5];
  for (int i = 0; i < 5; ++i) f[i] = (const float*)d_in[i];
  const float* attentions = (const float*)d_in[5];
  const float* key_t_w    = (const float*)d_in[6];
  const float* key_t_b    = (const float*)d_in[7];
  const float* value_t_w  = (const float*)d_in[8];
  const float* value_t_b  = (const float*)d_in[9];
  const float* key_q_w    = (const float*)d_in[10];
  const float* key_q_b    = (const float*)d_in[11];
  const float* value_q_w  = (const float*)d_in[12];
  const float* value_q_b  = (const float*)d_in[13];
  const float* bn_gamma   = (const float*)d_in[14];
  const float* bn_beta    = (const float*)d_in[15];
  const float* bn_mean    = (const float*)d_in[16];
  const float* bn_var     = (const float*)d_in[17];
  const float* combine_w  = (const float*)d_in[18];
  const float* combine_b  = (const float*)d_in[19];
  float* out = (float*)d_out;

  // workspace layout (floats), total 12,582,912 floats = 48 MB
  float* ws    = (float*)d_ws;
  float* FS    = ws;                   // 5 * 262144   downsampled feats
  float* KEYT  = FS    + 1310720;      // 163840       key_t  [20,32,256]
  float* VALT  = KEYT  + 163840;       // 655360       val_t  [20,128,256]
  float* VALTT = VALT  + 655360;       // 655360       val_t^T [20,256,128]
  float* KQ    = VALTT + 655360;       // 5 * 32768    kq
  float* VQ    = KQ    + 163840;       // 5 * 131072   vq
  float* VT    = VQ    + 655360;       // 5 * 131072   vt
  float* Pbuf  = VT    + 655360;       // 5242880      attn P [4,20,256,256]
  float* WTKT  = Pbuf  + 5242880;      // 98304
  float* WTVT  = WTKT  + 98304;        // 393216
  float* WTKQ  = WTVT  + 393216;       // 5 * 98304
  float* WTVQ  = WTKQ  + 491520;       // 5 * 393216
  float* WTC   = WTVQ  + 1966080;      // 131072

  static const int Hs[5] = {192, 96, 48, 24, 12};

  // 1. downsample all levels to 16x16
  for (int i = 0; i < 5; ++i)
    k_downsample<<<1024, 256, 0, stream>>>(f[i], FS + (size_t)i * 262144, Hs[i], Hs[i]);

  // 2. weight transposes
  k_tr_convw<<<(32 * 768 + 255) / 256, 256, 0, stream>>>(key_t_w, WTKT, 32);
  k_tr_convw<<<(128 * 768 + 255) / 256, 256, 0, stream>>>(value_t_w, WTVT, 128);
  for (int i = 0; i < 5; ++i) {
    k_tr_convw<<<(32 * 768 + 255) / 256, 256, 0, stream>>>(
        key_q_w + (size_t)i * 32 * 2304, WTKQ + (size_t)i * 98304, 32);
    k_tr_convw<<<(128 * 768 + 255) / 256, 256, 0, stream>>>(
        value_q_w + (size_t)i * 128 * 2304, WTVQ + (size_t)i * 393216, 128);
  }
  k_tr_combw<<<512, 256, 0, stream>>>(combine_w, WTC);

  // 3. template convs (on attentions) + val_t transpose
  k_conv3x3_wmma<32><<<dim3(20 * 8, 2), 32, 0, stream>>>(attentions, WTKT, key_t_b, KEYT);
  k_conv3x3_wmma<128><<<dim3(20 * 8, 8), 32, 0, stream>>>(attentions, WTVT, value_t_b, VALT);
  k_tr_valt<<<2560, 256, 0, stream>>>(VALT, VALTT);

  // 4. per-level query convs
  for (int i = 0; i < 5; ++i) {
    k_conv3x3_wmma<32><<<dim3(4 * 8, 2), 32, 0, stream>>>(
        FS + (size_t)i * 262144, WTKQ + (size_t)i * 98304, key_q_b + i * 32,
        KQ + (size_t)i * 32768);
    k_conv3x3_wmma<128><<<dim3(4 * 8, 8), 32, 0, stream>>>(
        FS + (size_t)i * 262144, WTVQ + (size_t)i * 393216, value_q_b + i * 128,
        VQ + (size_t)i * 131072);
  }

  // 5. per-level attention + fused combine (P reused; stream order serializes)
  size_t ooff = 0;
  for (int i = 0; i < 5; ++i) {
    int H = Hs[i], W = Hs[i];
    k_attn_logits<<<dim3(4, 20, 256), 32, 0, stream>>>(KQ + (size_t)i * 32768, KEYT, Pbuf);
    k_softmax_q<<<dim3(80, 16), 256, 0, stream>>>(Pbuf);
    k_attn_apply<<<dim3(4, 8, 16), 32, 0, stream>>>(VALTT, Pbuf, VT + (size_t)i * 131072);
    int ntile = (H * W + 31) / 32;
    k_combine<<<4 * ntile, 256, 0, stream>>>(f[i], WTC, combine_b,
        VQ + (size_t)i * 131072, VT + (size_t)i * 131072,
        bn_gamma + i * 256, bn_beta + i * 256, bn_mean + i * 256, bn_var + i * 256,
        out + ooff, H, W);
    ooff += (size_t)4 * 256 * H * W;
  }
}